// FocalLoss_45019847196876
// MI455X (gfx1250) — compile-verified
//
#include <hip/hip_runtime.h>
#include <math.h>

#define NB 8
#define NA 98304
#define NC 80
#define NM 32
#define FL_ALPHA 0.25f

typedef float v2f __attribute__((ext_vector_type(2)));
typedef float v8f __attribute__((ext_vector_type(8)));

// 16-byte payload type for the async b128 builtin (matches clang's prototype:
// 'int __attribute__((vector_size(16)))' pointers in AS1 (src) / AS3 (dst)).
typedef int b128vec __attribute__((vector_size(16)));
typedef __attribute__((address_space(1))) b128vec* gptr128;
typedef __attribute__((address_space(3))) b128vec* lptr128;

#if defined(__has_builtin)
#  if __has_builtin(__builtin_amdgcn_wmma_f32_16x16x4_f32)
#    define HAVE_WMMA_F32X4 1
#  endif
#  if __has_builtin(__builtin_amdgcn_global_load_async_to_lds_b128) && \
      __has_builtin(__builtin_amdgcn_s_wait_asynccnt)
#    define HAVE_ASYNC_LDS 1
#  endif
#endif

// ---- wave32 sum: matrix-pipe reduction via V_WMMA_F32_16X16X4_F32 ----
// A(16x4): lanes 0-15 VGPR0 = K0 rows 0-15, lanes 16-31 VGPR0 = K2 rows 0-15.
// With B = ones: D[m][*] = v_m + v_{m+16}. Per-lane sum of the 8 D VGPRs
// covers half the M rows; one xor-16 add completes the full 32-lane sum.
__device__ __forceinline__ float wave_sum32(float v) {
#ifdef HAVE_WMMA_F32X4
  v2f a; a[0] = v; a[1] = 0.0f;
  v2f bm; bm[0] = 1.0f; bm[1] = 1.0f;
  v8f c = {0.f, 0.f, 0.f, 0.f, 0.f, 0.f, 0.f, 0.f};
  c = __builtin_amdgcn_wmma_f32_16x16x4_f32(false, a, false, bm, (short)0, c,
                                            false, false);
  float s = c[0] + c[1] + c[2] + c[3] + c[4] + c[5] + c[6] + c[7];
  s += __shfl_xor(s, 16, 32);
  return s;
#else
  for (int off = 16; off > 0; off >>= 1) v += __shfl_xor(v, off, 32);
  return v;
#endif
}

__device__ __forceinline__ float smooth_l1(float x) {
  float d = fabsf(x);
  return (d <= (1.0f / 9.0f)) ? (4.5f * d * d) : (d - (1.0f / 18.0f));
}

// ---- zero the per-image accumulators (cls_sum[8], reg_sum[8], num_pos[8]) ----
__global__ void init_sums_kernel(float* sums) {
  if (threadIdx.x < 3 * NB) sums[threadIdx.x] = 0.0f;
}

// ---- anchor assignment: IoU argmax over M=32 GT (LDS), reg smooth-L1, state word ----
__global__ __launch_bounds__(256) void assign_kernel(
    const float* __restrict__ anchors, const float* __restrict__ regressions,
    const float* __restrict__ annotations, const float* __restrict__ neg_thr_p,
    const float* __restrict__ pos_thr_p, int* __restrict__ state,
    float* __restrict__ reg_sum, float* __restrict__ num_pos) {
  __shared__ float s_annot[NM * 5];
  __shared__ float s_red[16];
  const int b = blockIdx.y;
  const int tid = threadIdx.x;
  if (tid < NM * 5) s_annot[tid] = annotations[b * NM * 5 + tid];
  __syncthreads();

  const float neg_thr = *neg_thr_p;
  const float pos_thr = *pos_thr_p;
  const int a = blockIdx.x * 256 + tid;

  const float4 an = ((const float4*)anchors)[a];
  const float aw = an.z - an.x, ah = an.w - an.y;
  const float area_a = aw * ah;

  float best = -INFINITY;
  int arg = 0;
  for (int m = 0; m < NM; ++m) {
    float bx1 = s_annot[m * 5 + 0], by1 = s_annot[m * 5 + 1];
    float bx2 = s_annot[m * 5 + 2], by2 = s_annot[m * 5 + 3];
    float bcl = s_annot[m * 5 + 4];
    float area_b = (bx2 - bx1) * (by2 - by1);
    float iw = fmaxf(fminf(an.z, bx2) - fmaxf(an.x, bx1), 0.0f);
    float ih = fmaxf(fminf(an.w, by2) - fmaxf(an.y, by1), 0.0f);
    float inter = iw * ih;
    float ua = fmaxf(area_a + area_b - inter, 1e-8f);
    float iou = inter / ua;
    if (bcl == -1.0f) iou = -1.0f;  // invalid annotation
    if (iou > best) { best = iou; arg = m; }  // '>' keeps first argmax
  }

  const bool pos = best >= pos_thr;
  const bool neg = best < neg_thr;
  const float g0 = s_annot[arg * 5 + 0], g1 = s_annot[arg * 5 + 1];
  const float g2 = s_annot[arg * 5 + 2], g3 = s_annot[arg * 5 + 3];
  const int acls = (int)s_annot[arg * 5 + 4];

  state[(size_t)b * NA + a] = pos ? ((2 << 8) | acls) : (neg ? 0 : (1 << 8));

  // regression targets (gcx/gcy use pre-clip gw/gh, matching the reference)
  const float acx = an.x + 0.5f * aw, acy = an.y + 0.5f * ah;
  float gw = g2 - g0, gh = g3 - g1;
  const float gcx = g0 + 0.5f * gw, gcy = g1 + 0.5f * gh;
  gw = fmaxf(gw, 1.0f);
  gh = fmaxf(gh, 1.0f);
  const float t0 = ((gcx - acx) / aw) / 0.1f;
  const float t1 = ((gcy - acy) / ah) / 0.1f;
  const float t2 = logf(gw / aw) / 0.2f;
  const float t3 = logf(gh / ah) / 0.2f;
  const float4 r = ((const float4*)regressions)[(size_t)b * NA + a];
  const float sl = smooth_l1(t0 - r.x) + smooth_l1(t1 - r.y) +
                   smooth_l1(t2 - r.z) + smooth_l1(t3 - r.w);

  const float p_reg = pos ? sl : 0.0f;
  const float p_np = pos ? 1.0f : 0.0f;

  const float wr = wave_sum32(p_reg);  // all 256 lanes active: EXEC all-1s for WMMA
  const float wn = wave_sum32(p_np);
  const int lane = tid & 31, w = tid >> 5;
  if (lane == 0) { s_red[w] = wr; s_red[8 + w] = wn; }
  __syncthreads();
  if (tid == 0) {
    float tr = 0.f, tn = 0.f;
    for (int i = 0; i < 8; ++i) { tr += s_red[i]; tn += s_red[8 + i]; }
    atomicAdd(&reg_sum[b], tr);
    atomicAdd(&num_pos[b], tn);
  }
}

// ---- streaming classification focal loss: 252MB @ 23.3TB/s is the roofline ----
// Double-buffered GLOBAL_LOAD_ASYNC_TO_LDS_B128 pipeline; each lane reads back
// only its own 16B slot, so the hot loop needs no barriers, just ASYNCcnt.
__global__ __launch_bounds__(256) void cls_loss_kernel(
    const float* __restrict__ cls, const int* __restrict__ state,
    float* __restrict__ cls_sum) {
  __shared__ float4 buf[2][256];
  __shared__ float s_red[8];
  const int b = blockIdx.y;
  const int tid = threadIdx.x;
  const size_t img_base = (size_t)b * (size_t)NA * (size_t)NC;
  const int ntiles = (NA * NC) / 1024;  // 1024 floats (256 lanes x float4) per tile
  const int* st_img = state + (size_t)b * NA;
  float acc = 0.0f;

  auto process = [&](float4 v, unsigned e) {
    unsigned aidx = e / (unsigned)NC;  // NC=80, 80%4==0: float4 stays in-anchor
    unsigned c0 = e - aidx * (unsigned)NC;
    int stv = st_img[aidx];
    int flag = stv >> 8;  // 2=pos, 0=neg, 1=ignore
    if (flag != 1) {
      int acls = stv & 255;
      float pv[4] = {v.x, v.y, v.z, v.w};
#pragma unroll
      for (int j = 0; j < 4; ++j) {
        float p = fminf(fmaxf(pv[j], 1e-4f), 1.0f - 1e-4f);
        bool one = (flag == 2) && ((int)(c0 + j) == acls);
        float x = one ? (1.0f - p) : p;
        float af = one ? FL_ALPHA : (1.0f - FL_ALPHA);
        float bce = -logf(one ? p : (1.0f - p));
        acc += af * x * x * bce;
      }
    }
  };

#ifdef HAVE_ASYNC_LDS
  {
    const int t0 = (int)blockIdx.x;
    if (t0 < ntiles) {
      const float* g = cls + img_base + (size_t)t0 * 1024 + (size_t)tid * 4;
      __builtin_amdgcn_global_load_async_to_lds_b128(
          (gptr128)const_cast<float*>(g), (lptr128)&buf[0][tid], 0, 0);
    }
    int sb = 0;
    for (int t = t0; t < ntiles; t += (int)gridDim.x) {
      int tn = t + (int)gridDim.x;
      if (tn < ntiles) {
        const float* g = cls + img_base + (size_t)tn * 1024 + (size_t)tid * 4;
        __builtin_amdgcn_global_load_async_to_lds_b128(
            (gptr128)const_cast<float*>(g), (lptr128)&buf[sb ^ 1][tid], 0, 0);
        __builtin_amdgcn_s_wait_asynccnt(1);  // previous tile landed in LDS
      } else {
        __builtin_amdgcn_s_wait_asynccnt(0);
      }
      float4 v = buf[sb][tid];
      process(v, (unsigned)t * 1024u + (unsigned)tid * 4u);
      sb ^= 1;
    }
  }
#else
  for (int t = (int)blockIdx.x; t < ntiles; t += (int)gridDim.x) {
    float4 v = ((const float4*)(cls + img_base))[(size_t)t * 256 + tid];
    process(v, (unsigned)t * 1024u + (unsigned)tid * 4u);
  }
#endif

  const float ws = wave_sum32(acc);
  const int lane = tid & 31, w = tid >> 5;
  if (lane == 0) s_red[w] = ws;
  __syncthreads();
  if (tid == 0) {
    float tot = 0.f;
    for (int i = 0; i < 8; ++i) tot += s_red[i];
    atomicAdd(&cls_sum[b], tot);
  }
}

// ---- final per-image normalization + batch mean ----
__global__ void finalize_kernel(const float* __restrict__ cls_sum,
                                const float* __restrict__ reg_sum,
                                const float* __restrict__ num_pos,
                                float* __restrict__ out) {
  if (threadIdx.x == 0) {
    float cs = 0.f, rs = 0.f;
    for (int b = 0; b < NB; ++b) {
      float np = num_pos[b];
      cs += cls_sum[b] / fmaxf(np, 1.0f);
      rs += reg_sum[b] / fmaxf(np * 4.0f, 1.0f);
    }
    out[0] = cs / (float)NB;
    out[1] = rs / (float)NB;
  }
}

extern "C" void kernel_launch(void* const* d_in, const int* in_sizes, int n_in,
                              void* d_out, int out_size, void* d_ws,
                              size_t ws_size, hipStream_t stream) {
  (void)in_sizes; (void)n_in; (void)out_size; (void)ws_size;
  const float* cls     = (const float*)d_in[0];  // [B,A,C]
  const float* regs    = (const float*)d_in[1];  // [B,A,4]
  const float* anchors = (const float*)d_in[2];  // [1,A,4]
  const float* annots  = (const float*)d_in[3];  // [B,M,5]
  const float* neg_thr = (const float*)d_in[4];  // scalar
  const float* pos_thr = (const float*)d_in[5];  // scalar

  int* state  = (int*)d_ws;  // B*A ints
  float* sums = (float*)((char*)d_ws + (size_t)NB * NA * sizeof(int));
  float* cls_sum = sums;           // [NB]
  float* reg_sum = sums + NB;      // [NB]
  float* num_pos = sums + 2 * NB;  // [NB]

  init_sums_kernel<<<1, 32, 0, stream>>>(sums);

  dim3 g1(NA / 256, NB);
  assign_kernel<<<g1, 256, 0, stream>>>(anchors, regs, annots, neg_thr,
                                        pos_thr, state, reg_sum, num_pos);

  dim3 g2(768, NB);  // 6144 blocks; each streams ~10 x 4KB tiles per image slice
  cls_loss_kernel<<<g2, 256, 0, stream>>>(cls, state, cls_sum);

  finalize_kernel<<<1, 32, 0, stream>>>(cls_sum, reg_sum, num_pos,
                                        (float*)d_out);
}